// LengthRegulator_11879879543362
// MI455X (gfx1250) — compile-verified
//
#include <hip/hip_runtime.h>
#include <hip/hip_bf16.h>
#include <stdint.h>

// Problem constants (match reference)
#define BB   16
#define TT   1024
#define DD   384      // floats per token = 1536 bytes
#define TOUT 4096

// ---------------------------------------------------------------------------
// Kernel 1: counts = floor(max(dur,0)+0.5); csum = inclusive scan along T.
// 16 blocks x 1024 threads, Hillis-Steele scan in LDS. Negligible runtime.
// ---------------------------------------------------------------------------
__global__ __launch_bounds__(1024)
void lr_scan_kernel(const float* __restrict__ dur, int* __restrict__ csum) {
    __shared__ int s[TT];
    const int b = blockIdx.x;
    const int t = threadIdx.x;
    float d = dur[b * TT + t];
    int c = (int)floorf(fmaxf(d, 0.0f) + 0.5f);
    s[t] = c;
    __syncthreads();
#pragma unroll
    for (int off = 1; off < TT; off <<= 1) {
        int v = (t >= off) ? s[t - off] : 0;
        __syncthreads();
        s[t] += v;
        __syncthreads();
    }
    csum[b * TT + t] = s[t];
}

// ---------------------------------------------------------------------------
// Kernel 2: one wave32 per output frame.
//   idx = searchsorted_right(csum[b], j), clipped to T-1
//   out[b,j,:] = x[b,idx,:]   (1536 B copy via CDNA5 async global<->LDS DMA)
// Copy path: 3 x global_load_async_to_lds_b128 (512 B/wave each)
//            s_wait_asynccnt 0
//            3 x global_store_async_from_lds_b128
// ASYNCcnt-tracked; S_ENDPGM drains outstanding stores.
// ---------------------------------------------------------------------------
__global__ __launch_bounds__(256)
void lr_gather_kernel(const float* __restrict__ x,
                      const int* __restrict__ csum,
                      float* __restrict__ out) {
    __shared__ char lds_buf[8 * 1536];   // 1536 B staging region per wave

    const int wave = threadIdx.x >> 5;
    const int lane = threadIdx.x & 31;

    // frame id: 8 frames per block; TOUT % 8 == 0 so all waves of a block
    // share the same batch row b.
    const int frame = blockIdx.x * 8 + wave;
    int b = frame >> 12;        // / TOUT
    int j = frame & (TOUT - 1); // % TOUT

    // Uniform binary search (all lanes compute the same thing; loads
    // broadcast from the same address and the 1024-entry row lives in L0/L2).
    const int* __restrict__ c = csum + b * TT;
    int lo = 0, hi = TT;
#pragma unroll 1
    while (lo < hi) {
        int mid = (lo + hi) >> 1;
        if (c[mid] <= j) lo = mid + 1; else hi = mid;
    }
    int idx = (lo > TT - 1) ? (TT - 1) : lo;

    // Force wave-uniform values into SGPRs so the async ops can use the
    // SADDR (GVS) addressing mode with a scalar 64-bit base.
    int ub   = __builtin_amdgcn_readfirstlane(b);
    int uj   = __builtin_amdgcn_readfirstlane(j);
    int uidx = __builtin_amdgcn_readfirstlane(idx);

    const char* src = (const char*)(x   + ((size_t)ub * TT   + uidx) * DD);
    char*       dst = (char*)      (out + ((size_t)ub * TOUT + uj)   * DD);

    // LDS byte offset of this wave's staging region: the generic address of a
    // __shared__ object carries the LDS offset in its low 32 bits (the
    // addrspace(3)->generic cast builds {SHARED_BASE_HI, lds_offset}).
    unsigned lds0 = (unsigned)(uintptr_t)(&lds_buf[wave * 1536]);

    const unsigned lane_off = (unsigned)lane * 16u;   // b128 = 16 B per lane

    // global -> LDS (ASYNCcnt += 3)
#pragma unroll
    for (int k = 0; k < 3; ++k) {
        unsigned go = lane_off + (unsigned)k * 512u;  // global byte offset
        unsigned la = lds0 + go;                      // LDS byte address
        asm volatile("global_load_async_to_lds_b128 %0, %1, %2"
                     :
                     : "v"(la), "v"(go), "s"(src)
                     : "memory");
    }

    // Wait for the loads to land in LDS before the async stores read it.
    asm volatile("s_wait_asynccnt 0" ::: "memory");

    // LDS -> global (ASYNCcnt += 3); S_ENDPGM waits for completion.
#pragma unroll
    for (int k = 0; k < 3; ++k) {
        unsigned go = lane_off + (unsigned)k * 512u;
        unsigned la = lds0 + go;
        asm volatile("global_store_async_from_lds_b128 %0, %1, %2"
                     :
                     : "v"(go), "v"(la), "s"(dst)
                     : "memory");
    }
}

// ---------------------------------------------------------------------------
// Host-side launcher
//   d_in[0] = x   [16,1024,384] f32
//   d_in[1] = dur [16,1024]     f32
//   d_in[2] = out_len (scalar, fixed 4096 — hardcoded)
//   d_out   = out [16,4096,384] f32
//   d_ws    = csum [16,1024] i32 (64 KB)
// ---------------------------------------------------------------------------
extern "C" void kernel_launch(void* const* d_in, const int* in_sizes, int n_in,
                              void* d_out, int out_size, void* d_ws, size_t ws_size,
                              hipStream_t stream) {
    const float* x   = (const float*)d_in[0];
    const float* dur = (const float*)d_in[1];
    float* out       = (float*)d_out;
    int*   csum      = (int*)d_ws;

    lr_scan_kernel<<<dim3(BB), dim3(TT), 0, stream>>>(dur, csum);

    // 16*4096 frames, 8 waves (frames) per 256-thread block
    lr_gather_kernel<<<dim3((BB * TOUT) / 8), dim3(256), 0, stream>>>(x, csum, out);
}